// DecoderRNN_80504866997071
// MI455X (gfx1250) — compile-verified
//
#include <hip/hip_runtime.h>
#include <hip/hip_bf16.h>
#include <math.h>

typedef float v2f __attribute__((ext_vector_type(2)));
typedef float v8f __attribute__((ext_vector_type(8)));

#define Bsz 128
#define Pn  196
#define Ed  512
#define Hd  512
#define Ad  512
#define Vn  20000
#define Tn  24

// ---------------------------------------------------------------------------
// Generic fp32 WMMA GEMM:  C[M,N] = act(A[M,K] @ B + bias)
//   BT=false: B row-major [K,N], ldb = N   (element (k,n) = Bm[k*ldb + n])
//   BT=true : B stored    [N,K], ldb = K   (element (k,n) = Bm[n*ldb + k])
// One wave -> 16(M) x NT*16(N) strip via V_WMMA_F32_16X16X4_F32.
// Software-pipelined K loop (one 8-deep prefetch stage, A/B fragment sets
// ping-pong) so global-load latency overlaps the wmma bursts.
// K must be a multiple of 16 (here: 512 or 1024).
// ---------------------------------------------------------------------------
template <int ACT, int NT, bool BT>
__global__ __launch_bounds__(128) void wmma_gemm_f32(
    const float* __restrict__ A, const float* __restrict__ Bm,
    const float* __restrict__ bias, float* __restrict__ C,
    int M, int K, int lda, int ldb, long ldc, int n0base) {
  const int lane = threadIdx.x;    // 0..31
  const int half = lane >> 4;      // K sub-block selector (frag layout)
  const int lm   = lane & 15;      // row (A) / col (B) within tile
  const int m0   = (blockIdx.y * blockDim.y + threadIdx.y) * 16;
  if (m0 >= M) return;             // uniform per wave
  const int n0 = n0base + blockIdx.x * (NT * 16);

  v8f acc[NT] = {};
  const float* Arow = A + (long)(m0 + lm) * lda;
  const float* Brow[NT];
#pragma unroll
  for (int i = 0; i < NT; ++i)
    Brow[i] = BT ? (Bm + (long)(n0 + 16 * i + lm) * ldb)
                 : (Bm + (n0 + 16 * i + lm));

  auto load_frag = [&](int k, v2f& a0, v2f& a1, v2f (&b0)[NT], v2f (&b1)[NT]) {
    const int kk = k + 2 * half;
    a0 = *(const v2f*)(Arow + kk);
    a1 = *(const v2f*)(Arow + kk + 4);
#pragma unroll
    for (int i = 0; i < NT; ++i) {
      if (BT) {
        b0[i] = *(const v2f*)(Brow[i] + kk);
        b1[i] = *(const v2f*)(Brow[i] + kk + 4);
      } else {
        b0[i].x = Brow[i][(long)kk * ldb];
        b0[i].y = Brow[i][(long)(kk + 1) * ldb];
        b1[i].x = Brow[i][(long)(kk + 4) * ldb];
        b1[i].y = Brow[i][(long)(kk + 5) * ldb];
      }
    }
  };
  auto mma = [&](const v2f& a0, const v2f& a1, const v2f (&b0)[NT],
                 const v2f (&b1)[NT]) {
#pragma unroll
    for (int i = 0; i < NT; ++i)
      acc[i] = __builtin_amdgcn_wmma_f32_16x16x4_f32(
          false, a0, false, b0[i], (short)0, acc[i], false, false);
#pragma unroll
    for (int i = 0; i < NT; ++i)
      acc[i] = __builtin_amdgcn_wmma_f32_16x16x4_f32(
          false, a1, false, b1[i], (short)0, acc[i], false, false);
  };

  v2f a0A, a1A, a0B, a1B;
  v2f b0A[NT], b1A[NT], b0B[NT], b1B[NT];
  load_frag(0, a0A, a1A, b0A, b1A);
  int k = 0;
  for (; k + 16 < K; k += 16) {
    load_frag(k + 8, a0B, a1B, b0B, b1B);   // prefetch next half-chunk
    mma(a0A, a1A, b0A, b1A);                // overlaps set-B loads
    load_frag(k + 16, a0A, a1A, b0A, b1A);  // prefetch following chunk
    mma(a0B, a1B, b0B, b1B);                // overlaps set-A loads
  }
  load_frag(k + 8, a0B, a1B, b0B, b1B);
  mma(a0A, a1A, b0A, b1A);
  mma(a0B, a1B, b0B, b1B);

#pragma unroll
  for (int i = 0; i < NT; ++i) {
    const int n = n0 + 16 * i + lm;
    const float bv = bias ? bias[n] : 0.0f;
#pragma unroll
    for (int r = 0; r < 8; ++r) {
      float v = acc[i][r] + bv;
      if (ACT == 1) v = tanhf(v);
      C[(long)(m0 + r + 8 * half) * ldc + n] = v;
    }
  }
}

// avg[b,e] = mean_p features[b,p,e]
__global__ __launch_bounds__(256) void mean_kernel(const float* __restrict__ f,
                                                   float* __restrict__ avg) {
  int idx = blockIdx.x * blockDim.x + threadIdx.x;
  if (idx >= Bsz * Ed) return;
  int b = idx >> 9, e = idx & 511;
  const float* p = f + (long)b * Pn * Ed + e;
  float s = 0.f;
  for (int pp = 0; pp < Pn; ++pp) s += p[(long)pp * Ed];
  avg[idx] = s * (1.0f / (float)Pn);
}

// score[b,p] = sum_a tanh(feat_proj[b,p,a] + hWh[b,a]) * v_att[a]
__global__ __launch_bounds__(256) void attn_score_kernel(
    const float* __restrict__ fp, const float* __restrict__ hWh,
    const float* __restrict__ v_att, float* __restrict__ score) {
  int row = blockIdx.x * blockDim.y + threadIdx.y;  // b*Pn + p
  if (row >= Bsz * Pn) return;
  int b = row / Pn;
  int lane = threadIdx.x;
  const float* fr = fp + (long)row * Ad;
  const float* hr = hWh + (long)b * Ad;
  float acc = 0.f;
  for (int a = lane; a < Ad; a += 32) acc += tanhf(fr[a] + hr[a]) * v_att[a];
#pragma unroll
  for (int o = 16; o > 0; o >>= 1) acc += __shfl_xor(acc, o, 32);
  if (lane == 0) score[row] = acc;
}

// softmax over P per batch row
__global__ __launch_bounds__(256) void softmax_kernel(
    const float* __restrict__ score, float* __restrict__ alpha) {
  __shared__ float sh[256];
  int b = blockIdx.x, t = threadIdx.x;
  float v = (t < Pn) ? score[b * Pn + t] : -1e30f;
  sh[t] = v;
  __syncthreads();
  for (int s = 128; s > 0; s >>= 1) {
    if (t < s) sh[t] = fmaxf(sh[t], sh[t + s]);
    __syncthreads();
  }
  float mx = sh[0];
  __syncthreads();
  float e = (t < Pn) ? __expf(v - mx) : 0.0f;
  sh[t] = e;
  __syncthreads();
  for (int s = 128; s > 0; s >>= 1) {
    if (t < s) sh[t] += sh[t + s];
    __syncthreads();
  }
  float inv = 1.0f / sh[0];
  if (t < Pn) alpha[b * Pn + t] = e * inv;
}

// ctx[b,e] = sum_p alpha[b,p] * features[b,p,e]
__global__ __launch_bounds__(256) void ctx_kernel(const float* __restrict__ alpha,
                                                  const float* __restrict__ f,
                                                  float* __restrict__ ctx) {
  int idx = blockIdx.x * blockDim.x + threadIdx.x;
  if (idx >= Bsz * Ed) return;
  int b = idx >> 9, e = idx & 511;
  const float* fb = f + (long)b * Pn * Ed + e;
  const float* ab = alpha + b * Pn;
  float s = 0.f;
  for (int p = 0; p < Pn; ++p) s += ab[p] * fb[(long)p * Ed];
  ctx[idx] = s;
}

// x[b,:512] = (t==0 ? output0[b] : emb[captions[b,t-1]]),  x[b,512:] = ctx[b]
__global__ __launch_bounds__(256) void build_x_kernel(
    const float* __restrict__ out0, const float* __restrict__ emb,
    const int* __restrict__ captions, const float* __restrict__ ctx,
    float* __restrict__ x, int t) {
  int idx = blockIdx.x * blockDim.x + threadIdx.x;
  if (idx >= Bsz * 2 * Ed) return;
  int b = idx >> 10, i = idx & 1023;
  float v;
  if (i < Ed) {
    if (t == 0) v = out0[b * Ed + i];
    else {
      int tok = captions[b * Tn + (t - 1)];
      v = emb[(long)tok * Ed + i];
    }
  } else {
    v = ctx[b * Ed + (i - Ed)];
  }
  x[idx] = v;
}

__device__ __forceinline__ float sigmoidf_(float x) {
  return 1.0f / (1.0f + __expf(-x));
}

// GRU gates (PyTorch order r,z,n) -> h_new
__global__ __launch_bounds__(256) void gru_kernel(const float* __restrict__ gi,
                                                  const float* __restrict__ gh,
                                                  const float* __restrict__ h_old,
                                                  float* __restrict__ h_new) {
  int idx = blockIdx.x * blockDim.x + threadIdx.x;
  if (idx >= Bsz * Hd) return;
  int b = idx >> 9, j = idx & 511;
  const float* gib = gi + (long)b * 3 * Hd;
  const float* ghb = gh + (long)b * 3 * Hd;
  float r = sigmoidf_(gib[j] + ghb[j]);
  float z = sigmoidf_(gib[Hd + j] + ghb[Hd + j]);
  float n = tanhf(gib[2 * Hd + j] + r * ghb[2 * Hd + j]);
  h_new[idx] = (1.0f - z) * n + z * h_old[idx];
}

extern "C" void kernel_launch(void* const* d_in, const int* in_sizes, int n_in,
                              void* d_out, int out_size, void* d_ws, size_t ws_size,
                              hipStream_t stream) {
  (void)in_sizes; (void)n_in; (void)out_size; (void)ws_size;
  const float* features = (const float*)d_in[0];
  const int*   captions = (const int*)d_in[1];
  const float* emb      = (const float*)d_in[2];
  const float* W_init_h = (const float*)d_in[3];
  const float* b_init_h = (const float*)d_in[4];
  const float* W_init_o = (const float*)d_in[5];
  const float* b_init_o = (const float*)d_in[6];
  const float* Wf       = (const float*)d_in[7];
  const float* bf       = (const float*)d_in[8];
  const float* Wh       = (const float*)d_in[9];
  const float* bh       = (const float*)d_in[10];
  const float* v_att    = (const float*)d_in[11];
  const float* W_ih     = (const float*)d_in[12];
  const float* W_hh     = (const float*)d_in[13];
  const float* b_ih     = (const float*)d_in[14];
  const float* b_hh     = (const float*)d_in[15];
  const float* W_out    = (const float*)d_in[16];
  const float* b_out    = (const float*)d_in[17];
  float* out = (float*)d_out;

  // workspace layout (floats)
  float* ws = (float*)d_ws;
  float* feat_proj = ws; ws += (long)Bsz * Pn * Ad;   // 12.85M
  float* avg   = ws; ws += Bsz * Ed;
  float* h0    = ws; ws += Bsz * Hd;
  float* h1    = ws; ws += Bsz * Hd;
  float* out0  = ws; ws += Bsz * Hd;
  float* hWh   = ws; ws += Bsz * Ad;
  float* score = ws; ws += Bsz * Pn;
  float* alpha = ws; ws += Bsz * Pn;
  float* ctxb  = ws; ws += Bsz * Ed;
  float* xb    = ws; ws += Bsz * 2 * Ed;
  float* gi    = ws; ws += Bsz * 3 * Hd;
  float* gh    = ws; ws += Bsz * 3 * Hd;

  const dim3 gblk(32, 4);  // 4 waves/block -> 64 rows per block
  auto ggrid = [](int M, int Nfull) {
    return dim3(Nfull / 64, (M + 63) / 64);
  };

  // ---- init: avg, hidden0, output0, feat_proj ----
  mean_kernel<<<(Bsz * Ed + 255) / 256, 256, 0, stream>>>(features, avg);
  wmma_gemm_f32<1, 4, false><<<ggrid(Bsz, Hd), gblk, 0, stream>>>(
      avg, W_init_h, b_init_h, h0, Bsz, Ed, Ed, Hd, Hd, 0);
  wmma_gemm_f32<1, 4, false><<<ggrid(Bsz, Hd), gblk, 0, stream>>>(
      avg, W_init_o, b_init_o, out0, Bsz, Ed, Ed, Hd, Hd, 0);
  wmma_gemm_f32<0, 4, false><<<ggrid(Bsz * Pn, Ad), gblk, 0, stream>>>(
      features, Wf, bf, feat_proj, Bsz * Pn, Ed, Ed, Ad, Ad, 0);

  const int VnFull = (Vn / 64) * 64;  // 19968 (312 strips); tail = 32 cols (NT=2)
  float* hcur = h0;
  float* hnext = h1;
  for (int t = 0; t < Tn; ++t) {
    // attention
    wmma_gemm_f32<0, 4, false><<<ggrid(Bsz, Ad), gblk, 0, stream>>>(
        hcur, Wh, bh, hWh, Bsz, Hd, Hd, Ad, Ad, 0);
    attn_score_kernel<<<(Bsz * Pn + 7) / 8, dim3(32, 8), 0, stream>>>(
        feat_proj, hWh, v_att, score);
    softmax_kernel<<<Bsz, 256, 0, stream>>>(score, alpha);
    ctx_kernel<<<(Bsz * Ed + 255) / 256, 256, 0, stream>>>(alpha, features, ctxb);
    // GRU input
    build_x_kernel<<<(Bsz * 2 * Ed + 255) / 256, 256, 0, stream>>>(
        out0, emb, captions, ctxb, xb, t);
    // gi = x @ W_ih^T + b_ih   (W_ih stored [3H, 2E] -> BT, ldb = 2E)
    wmma_gemm_f32<0, 4, true><<<ggrid(Bsz, 3 * Hd), gblk, 0, stream>>>(
        xb, W_ih, b_ih, gi, Bsz, 2 * Ed, 2 * Ed, 2 * Ed, 3 * Hd, 0);
    // gh = h @ W_hh^T + b_hh   (W_hh stored [3H, H] -> BT, ldb = H)
    wmma_gemm_f32<0, 4, true><<<ggrid(Bsz, 3 * Hd), gblk, 0, stream>>>(
        hcur, W_hh, b_hh, gh, Bsz, Hd, Hd, Hd, 3 * Hd, 0);
    gru_kernel<<<(Bsz * Hd + 255) / 256, 256, 0, stream>>>(gi, gh, hcur, hnext);
    // pred: h_new @ W_out^T + b_out -> out[:, t, :]  (ldc = T*V)
    wmma_gemm_f32<0, 4, true><<<ggrid(Bsz, VnFull), gblk, 0, stream>>>(
        hnext, W_out, b_out, out + (long)t * Vn, Bsz, Hd, Hd, Hd,
        (long)Tn * Vn, 0);
    // vocab tail: columns [19968, 20000) -> 2 tiles of 16
    wmma_gemm_f32<0, 2, true><<<dim3(1, 2), gblk, 0, stream>>>(
        hnext, W_out, b_out, out + (long)t * Vn, Bsz, Hd, Hd, Hd,
        (long)Tn * Vn, VnFull);
    float* tmp = hcur; hcur = hnext; hnext = tmp;
  }
}